// LDRLDR2_50431505989650
// MI455X (gfx1250) — compile-verified
//
#include <hip/hip_runtime.h>

typedef float v2f __attribute__((ext_vector_type(2)));
typedef float v8f __attribute__((ext_vector_type(8)));
typedef unsigned int u32x4 __attribute__((ext_vector_type(4)));
typedef int i32x4 __attribute__((ext_vector_type(4)));
typedef int i32x8 __attribute__((ext_vector_type(8)));

#define NCH 4096
#define NIN 1024
#define BSZ 128
#define FCW 512

#define KC   64   // K-depth of each LDS-staged W tile
#define NW   32   // N-width of each block's strip (2 WMMA tiles)
#define LDSW 68   // padded LDS row stride in dwords (64 + 4 TDM pad) -> 64-bank conflict-free

// ---- c = [1, cumprod(subd_A)], d = [1, cumprod(subd_B)] ----
__global__ void cumprod_kernel(const float* __restrict__ sA,
                               const float* __restrict__ sB,
                               float* __restrict__ c, float* __restrict__ d) {
  int t = threadIdx.x;
  if (t == 0) {
    float acc = 1.f; c[0] = 1.f;
    for (int i = 1; i < NCH; ++i) { acc *= sA[i - 1]; c[i] = acc; }
  } else if (t == 1) {
    float acc = 1.f; d[0] = 1.f;
    for (int i = 1; i < NCH; ++i) { acc *= sB[i - 1]; d[i] = acc; }
  }
}

// gpT[a*R + r] = G[r,a]/c[a];  q[r,a] = H[r,a]/d[a]
__global__ void prep_kernel(const float* __restrict__ G, const float* __restrict__ H,
                            const float* __restrict__ c, const float* __restrict__ d,
                            float* __restrict__ gpT, float* __restrict__ q, int R) {
  int e = blockIdx.x * blockDim.x + threadIdx.x;
  if (e >= R * NCH) return;
  int a = e & (NCH - 1);
  int r = e >> 12;
  gpT[a * R + r] = G[e] / c[a];
  q[e] = H[e] / d[a];
}

// T[a,b] = sum_r gpT[a,r] * q[r,b] ; one 16x16 tile per wave, f32 WMMA K=4
__global__ void gemmT_kernel(const float* __restrict__ gpT, const float* __restrict__ q,
                             float* __restrict__ T, int R) {
  int wave = (blockIdx.x * blockDim.x + threadIdx.x) >> 5;
  int lane = threadIdx.x & 31;
  int hf = lane >> 4, l = lane & 15;
  int a0 = (wave >> 8) * 16;
  int b0 = (wave & 255) * 16;
  v8f acc = {};
  for (int kk = 0; kk < R; kk += 4) {
    v2f af = *(const v2f*)(gpT + (size_t)(a0 + l) * R + kk + 2 * hf);
    v2f bf;
    bf.x = q[(size_t)(kk + 2 * hf + 0) * NCH + b0 + l];
    bf.y = q[(size_t)(kk + 2 * hf + 1) * NCH + b0 + l];
    acc = __builtin_amdgcn_wmma_f32_16x16x4_f32(false, af, false, bf,
                                                (short)0, acc, false, false);
  }
#pragma unroll
  for (int j = 0; j < 8; ++j)
    T[(size_t)(a0 + j + 8 * hf) * NCH + b0 + l] = acc[j];
}

// In-place: S[i,j] = T[i,j] + S[i-1,j-1]; then W[i,j] = c[i]*d[j]*S[i,j]
__global__ void diagscan_kernel(float* __restrict__ T, const float* __restrict__ c,
                                const float* __restrict__ dv) {
  int t = blockIdx.x * blockDim.x + threadIdx.x;
  if (t >= 2 * NCH - 1) return;
  int off = t - (NCH - 1);  // j - i
  float s = 0.f;
  for (int i = 0; i < NCH; ++i) {
    int j = i + off;
    if ((unsigned)j < (unsigned)NCH) {
      size_t idx = (size_t)i * NCH + j;
      s += T[idx];
      T[idx] = c[i] * dv[j] * s;
    }
  }
}

// Zero-pad x (128 x 1024) into xp (128 x 4096)
__global__ void pad_kernel(const float* __restrict__ x, float* __restrict__ xp) {
  int e = blockIdx.x * blockDim.x + threadIdx.x;
  int b = e >> 12, i = e & (NCH - 1);
  xp[e] = (i < NIN) ? x[b * NIN + i] : 0.f;
}

// out[b,i] = relu( sum_m in[b,m] * W[i,m] + bias[i] )
// Block = 8 waves covering all 128 batch rows for one 32-wide N strip.
// W K-slices (32 rows x 64 cols) staged into LDS by the Tensor Data Mover
// (double-buffered, TENSORcnt-synced) and broadcast to all 8 waves.
__global__ void gemm_y_kernel(const float* __restrict__ in, const float* __restrict__ W,
                              const float* __restrict__ bias, float* __restrict__ out) {
  __shared__ float smem[2][NW * LDSW];
  int wid  = threadIdx.x >> 5;
  int lane = threadIdx.x & 31;
  int hf = lane >> 4, l = lane & 15;
  int n0 = blockIdx.x * NW;   // N strip
  int b0 = wid * 16;          // M tile (batch rows)

  unsigned long long wbase = (unsigned long long)(uintptr_t)(W + (size_t)n0 * NCH);
  unsigned int lds[2];
  lds[0] = (unsigned int)(uintptr_t)&smem[0][0];
  lds[1] = (unsigned int)(uintptr_t)&smem[1][0];

  // D# group1: wg_mask=0 | data_size=4B | pad_en | pad_interval=64dw | pad_amount=4dw ;
  // tensor_dim0=4096, tensor_dim1=4096, tile_dim0=64(K), tile_dim1=32(N rows),
  // tensor_dim0_stride=4096.
  const i32x8 g1 = { 0x07520000, 0x10000000, 0x10000000, 0x00400000,
                     NW, NCH, 0, 0 };
  const i32x4 gz4 = { 0, 0, 0, 0 };
  const i32x8 gz8 = { 0, 0, 0, 0, 0, 0, 0, 0 };

  auto issue = [&](int chunk, unsigned int ldsaddr) {
    unsigned long long ga = wbase + (unsigned long long)chunk * (KC * 4);
    u32x4 g0;
    g0.x = 1u;                                   // count=1 valid descriptor
    g0.y = ldsaddr;                              // lds_addr (bytes)
    g0.z = (unsigned int)ga;                     // global_addr[31:0]
    g0.w = (unsigned int)(ga >> 32) | (2u << 30);// global_addr[56:32] | type=2
    __builtin_amdgcn_tensor_load_to_lds(g0, g1, gz4, gz4, gz8, 0);
  };

  const float* arow = in + (size_t)(b0 + l) * NCH + 2 * hf;
  v8f acc0 = {}, acc1 = {};

  if (wid == 0) issue(0, lds[0]);

  for (int c = 0; c < NCH / KC; ++c) {
    if (wid == 0) {
      if (c + 1 < NCH / KC) {
        issue(c + 1, lds[(c + 1) & 1]);
        __builtin_amdgcn_s_wait_tensorcnt(1);   // chunk c complete (TDM in-order)
      } else {
        __builtin_amdgcn_s_wait_tensorcnt(0);
      }
    }
    __syncthreads();
    const float* bb = smem[c & 1] + 2 * hf;
    int mbase = c * KC;
#pragma unroll
    for (int mm = 0; mm < KC; mm += 4) {
      v2f af  = *(const v2f*)(arow + mbase + mm);
      v2f bf0 = *(const v2f*)(bb + (0 * 16 + l) * LDSW + mm);
      v2f bf1 = *(const v2f*)(bb + (1 * 16 + l) * LDSW + mm);
      acc0 = __builtin_amdgcn_wmma_f32_16x16x4_f32(false, af, false, bf0, (short)0, acc0, false, false);
      acc1 = __builtin_amdgcn_wmma_f32_16x16x4_f32(false, af, false, bf1, (short)0, acc1, false, false);
    }
    __syncthreads();
  }

#pragma unroll
  for (int j = 0; j < 8; ++j) {
    int row = b0 + j + 8 * hf;
    float v;
    v = acc0[j] + bias[n0 + l];      out[(size_t)row * NCH + n0 + l]      = v > 0.f ? v : 0.f;
    v = acc1[j] + bias[n0 + 16 + l]; out[(size_t)row * NCH + n0 + 16 + l] = v > 0.f ? v : 0.f;
  }
}

// logits: out[b,o] = b_logits[o] + sum_{k<512} h[b,k] * Wl[o,k]
__global__ void logits_kernel(const float* __restrict__ h, const float* __restrict__ Wl,
                              const float* __restrict__ bl, float* __restrict__ out) {
  int e = blockIdx.x * blockDim.x + threadIdx.x;
  if (e >= BSZ * 10) return;
  int b = e / 10, o = e % 10;
  float s = bl[o];
  const float* hr = h + (size_t)b * NCH;
  const float* wr = Wl + o * FCW;
  for (int k = 0; k < FCW; ++k) s += hr[k] * wr[k];
  out[e] = s;
}

extern "C" void kernel_launch(void* const* d_in, const int* in_sizes, int n_in,
                              void* d_out, int out_size, void* d_ws, size_t ws_size,
                              hipStream_t stream) {
  (void)in_sizes; (void)n_in; (void)out_size; (void)ws_size;
  const float* x   = (const float*)d_in[0];
  const float* G1  = (const float*)d_in[1];
  const float* H1  = (const float*)d_in[2];
  const float* sA1 = (const float*)d_in[3];
  const float* sB1 = (const float*)d_in[4];
  const float* b1  = (const float*)d_in[5];
  const float* G2  = (const float*)d_in[6];
  const float* H2  = (const float*)d_in[7];
  const float* sA2 = (const float*)d_in[8];
  const float* sB2 = (const float*)d_in[9];
  const float* b2  = (const float*)d_in[10];
  const float* Wl  = (const float*)d_in[11];
  const float* bl  = (const float*)d_in[12];
  float* out = (float*)d_out;

  float* ws = (float*)d_ws;
  float* cA  = ws;  ws += 4096;
  float* cB  = ws;  ws += 4096;
  float* gpT = ws;  ws += 48 * NCH;
  float* qm  = ws;  ws += 48 * NCH;
  float* xp  = ws;  ws += BSZ * NCH;
  float* h1  = ws;  ws += BSZ * NCH;
  float* h2  = ws;  ws += BSZ * NCH;
  float* T   = ws;  // 4096*4096 floats (64 MB)

  dim3 blk(256);
  // ---- layer 1 (rank 48) ----
  cumprod_kernel<<<1, 64, 0, stream>>>(sA1, sB1, cA, cB);
  prep_kernel<<<(48 * NCH + 255) / 256, blk, 0, stream>>>(G1, H1, cA, cB, gpT, qm, 48);
  gemmT_kernel<<<8192, blk, 0, stream>>>(gpT, qm, T, 48);
  diagscan_kernel<<<(2 * NCH + 255) / 256, blk, 0, stream>>>(T, cA, cB);
  pad_kernel<<<(BSZ * NCH) / 256, blk, 0, stream>>>(x, xp);
  gemm_y_kernel<<<NCH / NW, blk, 0, stream>>>(xp, T, b1, h1);   // 128 blocks x 8 waves
  // ---- layer 2 (rank 16) ----
  cumprod_kernel<<<1, 64, 0, stream>>>(sA2, sB2, cA, cB);
  prep_kernel<<<(16 * NCH + 255) / 256, blk, 0, stream>>>(G2, H2, cA, cB, gpT, qm, 16);
  gemmT_kernel<<<8192, blk, 0, stream>>>(gpT, qm, T, 16);
  diagscan_kernel<<<(2 * NCH + 255) / 256, blk, 0, stream>>>(T, cA, cB);
  gemm_y_kernel<<<NCH / NW, blk, 0, stream>>>(h1, T, b2, h2);
  // ---- logits ----
  logits_kernel<<<(BSZ * 10 + 255) / 256, blk, 0, stream>>>(h2, Wl, bl, out);
}